// NH_PE_61177514164980
// MI455X (gfx1250) — compile-verified
//
#include <hip/hip_runtime.h>
#include <hip/hip_bf16.h>

// Neo-Hookean plane-strain point model, fully closed-form (no jacfwd needed).
// Bandwidth-bound: ~100 B/point, ~4.3 us at 23.3 TB/s HBM peak.
// CDNA5 paths: global_load_async_to_lds_b128 (ASYNCcnt) double-buffered
// pipeline, s_wait_asynccnt, wave32-sized blocks, b128 coalesced stores.

#define TILE 256

__global__ __launch_bounds__(TILE) void nh_pe_kernel(
    const float4* __restrict__ F_in, const float* __restrict__ mat_par,
    float* __restrict__ out, int N, int nTiles)
{
  __shared__ float4 sF[2][TILE];
  const int tid = (int)threadIdx.x;
  const float c1 = mat_par[0];   // uniform -> scalar loads
  const float c2 = mat_par[1];

  float*  outPsi = out;                                        // [N]
  float4* outC   = reinterpret_cast<float4*>(out + (size_t)N); // [N]
  float4* outD   = reinterpret_cast<float4*>(out + (size_t)5 * (size_t)N); // [N*4]

  int t   = (int)blockIdx.x;
  int buf = 0;

  // ---- prime the pipeline: async-stage tile t into LDS buffer 0 ----
  if (t < nTiles) {
    int gi = t * TILE + tid; if (gi >= N) gi = N - 1;   // clamp OOB reads
    unsigned ldsA = (unsigned)(size_t)&sF[0][tid];      // low 32b of LDS aperture addr = LDS offset
    unsigned gOff = (unsigned)gi * 16u;
    asm volatile("global_load_async_to_lds_b128 %0, %1, %2"
                 :: "v"(ldsA), "v"(gOff), "s"(F_in) : "memory");
  }

  for (; t < nTiles; t += (int)gridDim.x) {
    // Wait for this wave's staged tile. Each LDS slot is produced and
    // consumed by the same lane, so no cross-wave barrier is required.
    asm volatile("s_wait_asynccnt 0x0" ::: "memory");

    // Prefetch the next tile into the other buffer while we compute.
    int nxt = t + (int)gridDim.x;
    if (nxt < nTiles) {
      int gi = nxt * TILE + tid; if (gi >= N) gi = N - 1;
      unsigned ldsA = (unsigned)(size_t)&sF[buf ^ 1][tid];
      unsigned gOff = (unsigned)gi * 16u;
      asm volatile("global_load_async_to_lds_b128 %0, %1, %2"
                   :: "v"(ldsA), "v"(gOff), "s"(F_in) : "memory");
    }

    float4 f = sF[buf][tid];            // ds_load_b128
    float a = f.x, b = f.y, c = f.z, d = f.w;   // F row-major

    float J   = a * d - b * c;
    float B00 = a * a + b * b;          // B = F F^T
    float B11 = c * c + d * d;
    float B01 = a * c + b * d;
    float tr  = B00 + B11;
    float I1  = tr + 1.0f;              // trace(F^T F) + 1

    float rJ   = __builtin_amdgcn_rcpf(J);
    float Jm23 = __builtin_amdgcn_exp2f(-0.66666666667f * __builtin_amdgcn_logf(J)); // J^(-2/3)

    float alpha = 2.0f * c1 * Jm23;
    float Jm1   = J - 1.0f;
    float gamma = -alpha * I1 * (1.0f / 3.0f) + 2.0f * c2 * J * Jm1;
    float c3    = (2.0f / 9.0f) * alpha * I1 + 2.0f * c2 * J * (2.0f * J - 1.0f);

    float psi = c1 * Jm23 * I1 - 3.0f + c2 * Jm1 * Jm1;

    // Kirchhoff K = P F^T = alpha*B + gamma*I ; Cauchy = K / J
    float K00 = alpha * B00 + gamma;
    float K11 = alpha * B11 + gamma;
    float K01 = alpha * B01;
    float C00 = K00 * rJ, C11 = K11 * rJ, C01 = K01 * rJ;

    // Closed-form spatial tangent in Voigt slots (0,0),(1,1),(0,1):
    float a23 = (2.0f / 3.0f) * alpha;
    float D00 = (a23 * B00 + c3) * rJ;
    float D11 = (a23 * B11 + c3) * rJ;
    float D01 = (c3 - a23 * tr) * rJ;
    float Dof = ((1.0f / 3.0f) * alpha * B01) * rJ;   // D02=D12=D20=D21
    float D22 = 0.5f * alpha * tr * rJ;

    int i = t * TILE + tid;
    if (i < N) {
      outPsi[i] = psi;
      outC[i]   = make_float4(C00, C11, 0.0f, C01);
      float4* Dp = outD + (size_t)i * 4;                 // 4x b128 stores
      Dp[0] = make_float4(D00, D01, 0.0f, Dof);
      Dp[1] = make_float4(D01, D11, 0.0f, Dof);
      Dp[2] = make_float4(0.0f, 0.0f, 1.0f, 0.0f);
      Dp[3] = make_float4(Dof, Dof, 0.0f, D22);
    }
    buf ^= 1;
  }
}

extern "C" void kernel_launch(void* const* d_in, const int* in_sizes, int n_in,
                              void* d_out, int out_size, void* d_ws, size_t ws_size,
                              hipStream_t stream) {
  (void)n_in; (void)out_size; (void)d_ws; (void)ws_size;
  const float4* F  = (const float4*)d_in[0];   // (N,2,2) f32 row-major = float4/point
  const float*  mp = (const float*)d_in[1];    // (2,) f32
  float* out = (float*)d_out;                  // [psi N | Cauchy4 4N | D44 16N]
  int N = in_sizes[0] / 4;
  int nTiles = (N + TILE - 1) / TILE;
  int grid = nTiles < 1024 ? nTiles : 1024;    // persistent blocks -> pipelined tiles
  nh_pe_kernel<<<grid, TILE, 0, stream>>>(F, mp, out, N, nTiles);
}